// CGpool_27195732918501
// MI455X (gfx1250) — compile-verified
//
#include <hip/hip_runtime.h>
#include <hip/hip_bf16.h>

// ---------------------------------------------------------------------------
// Sizes (match reference)
// ---------------------------------------------------------------------------
#define N_CONV   3
#define DIM      128
#define N_CGS    8
#define N_ATOMS  65536
#define N_MOLS   128
#define N_PER    512           // N_ATOMS / N_MOLS
#define E_BOND   131072
#define E_EDGE   65536
#define BC       1024          // N_MOLS * N_CGS

typedef __attribute__((ext_vector_type(2))) float v2f;
typedef __attribute__((ext_vector_type(8))) float v8f;

// ---------------------------------------------------------------------------
// Embedding gather: h[i,:] = (node==0) ? 0 : embed[node,:]
// thread = (atom, 4-float chunk)
// ---------------------------------------------------------------------------
__global__ __launch_bounds__(256) void k_gather(const int* __restrict__ nodes,
                                                const float* __restrict__ embed,
                                                float* __restrict__ H)
{
    int tid  = blockIdx.x * 256 + threadIdx.x;
    int atom = tid >> 5;
    int c4   = (tid & 31) * 4;
    if (atom >= N_ATOMS) return;
    int idx = nodes[atom];
    float4 v = make_float4(0.f, 0.f, 0.f, 0.f);
    if (idx != 0)
        v = *(const float4*)(embed + (size_t)idx * DIM + c4);
    *(float4*)(H + (size_t)atom * DIM + c4) = v;
}

// ---------------------------------------------------------------------------
// WMMA fp32 GEMM: Out[M x 128] = act(A[M x 128] @ W[128 x 128] + bias)
// One workgroup = 8 waves = 128 rows. W staged in LDS (64 KB), swizzled as
// K-pairs so each B fragment is ONE aligned ds_load_b64:
//   ldsW2[(k>>1)*128 + col] = { W[k][col], W[k+1][col] }
//
// Fragment layout (CDNA5 ISA 7.12.2, 32-bit 16x4 matrices):
//   A: lane L holds row m=L%16; VGPR v holds K = v + 2*(L/16)  -> float2/lane
//   B: lane L holds col n=L%16; VGPR v holds K = v + 2*(L/16)
//   C/D: VGPR j, lane L -> element (M = j + 8*(L/16), N = L%16)
//
// In-place (Out == A) is safe: each row strip is read only by the wave that
// writes it, and all its reads happen before any write.
// ---------------------------------------------------------------------------
__global__ __launch_bounds__(256) void k_gemm128(const float* __restrict__ A,
                                                 const float* __restrict__ W,
                                                 const float* __restrict__ bias,
                                                 float* __restrict__ Out,
                                                 int M, int do_tanh)
{
    __shared__ v2f ldsW2[(DIM / 2) * DIM];       // 64 KB, K-pair swizzled
    // Stage: thread handles (kpair p, col c); reads two strided rows (L2-hot),
    // writes one contiguous b64.
    for (int i = threadIdx.x; i < (DIM / 2) * DIM; i += 256) {
        int p = i >> 7;                          // k pair index, 0..63
        int c = i & 127;                         // column
        v2f w;
        w.x = W[(2 * p)     * DIM + c];
        w.y = W[(2 * p + 1) * DIM + c];
        ldsW2[i] = w;
    }
    __syncthreads();

    const int wave = threadIdx.x >> 5;
    const int lane = threadIdx.x & 31;
    const int m    = lane & 15;
    const int hi   = lane >> 4;                  // 0 or 1
    const int row0 = (blockIdx.x * 8 + wave) * 16;
    if (row0 >= M) return;                       // uniform per wave (EXEC all-1s for WMMA)

    v8f acc[8];
#pragma unroll
    for (int t = 0; t < 8; ++t) acc[t] = (v8f){};

    const float* Arow = A + (size_t)(row0 + m) * DIM;

#pragma unroll 4
    for (int kb = 0; kb < DIM; kb += 4) {
        // A fragment: contiguous float2, 8-byte aligned
        v2f a = *(const v2f*)(Arow + kb + 2 * hi);
        const v2f* brow = ldsW2 + ((kb >> 1) + hi) * DIM + m;
#pragma unroll
        for (int t = 0; t < 8; ++t) {
            v2f b = brow[t * 16];                // one ds_load_b64
            acc[t] = __builtin_amdgcn_wmma_f32_16x16x4_f32(
                false, a, false, b, (short)0, acc[t], false, false);
        }
    }

#pragma unroll
    for (int t = 0; t < 8; ++t) {
        const int c0 = t * 16;
        const float bv = bias[c0 + m];
#pragma unroll
        for (int j = 0; j < 8; ++j) {
            const int r = row0 + j + 8 * hi;
            float v = acc[t][j] + bv;
            if (do_tanh) v = tanhf(v);
            Out[(size_t)r * DIM + c0 + m] = v;
        }
    }
}

// ---------------------------------------------------------------------------
// Bidirectional bond scatter: for each bond (a,b): H[a]+=Msg[b], H[b]+=Msg[a]
// thread = (directed edge, 4-float chunk)
// ---------------------------------------------------------------------------
__global__ __launch_bounds__(256) void k_scatter(const int* __restrict__ bonds,
                                                 const float* __restrict__ Msg,
                                                 float* __restrict__ H)
{
    int tid = blockIdx.x * 256 + threadIdx.x;
    int e   = tid >> 5;                           // directed edge id
    int c4  = (tid & 31) * 4;
    if (e >= 2 * E_BOND) return;
    int dst, src;
    if (e < E_BOND) { dst = bonds[2 * e];                 src = bonds[2 * e + 1]; }
    else            { int f = e - E_BOND; dst = bonds[2 * f + 1]; src = bonds[2 * f]; }
    float4 v = *(const float4*)(Msg + (size_t)src * DIM + c4);
    float* d = H + (size_t)dst * DIM + c4;
    atomicAdd(d + 0, v.x);
    atomicAdd(d + 1, v.y);
    atomicAdd(d + 2, v.z);
    atomicAdd(d + 3, v.w);
}

// ---------------------------------------------------------------------------
// Head: hl = T @ Wc2 + bc2  (128 -> 8), then softmax -> a. One row per thread.
// ---------------------------------------------------------------------------
__global__ __launch_bounds__(256) void k_head(const float* __restrict__ T,
                                              const float* __restrict__ Wc2,
                                              const float* __restrict__ bc2,
                                              float* __restrict__ hl_out,
                                              float* __restrict__ a_out)
{
    int row = blockIdx.x * 256 + threadIdx.x;
    if (row >= N_ATOMS) return;
    float acc[N_CGS];
#pragma unroll
    for (int c = 0; c < N_CGS; ++c) acc[c] = bc2[c];
    const float* tr = T + (size_t)row * DIM;
    for (int k = 0; k < DIM; ++k) {
        float tv = tr[k];
        const float* wr = Wc2 + k * N_CGS;        // row-major (D, N_CGS)
#pragma unroll
        for (int c = 0; c < N_CGS; ++c) acc[c] += tv * wr[c];
    }
    float mx = acc[0];
#pragma unroll
    for (int c = 1; c < N_CGS; ++c) mx = fmaxf(mx, acc[c]);
    float s = 0.f, ex[N_CGS];
#pragma unroll
    for (int c = 0; c < N_CGS; ++c) { ex[c] = __expf(acc[c] - mx); s += ex[c]; }
    float inv = 1.f / s;
#pragma unroll
    for (int c = 0; c < N_CGS; ++c) {
        hl_out[(size_t)row * N_CGS + c] = acc[c];
        a_out [(size_t)row * N_CGS + c] = ex[c] * inv;
    }
}

// ---------------------------------------------------------------------------
// Per-mol reduction (deterministic, no atomics):
//   colsum[mol*8+c]        = sum_{atoms in mol} a[atom,c]
//   cg_xyz[(mol*8+c),k]    = sum_{atoms} a[atom,c]*xyz[atom,k] / colsum
// One block per mol; 32 partials per thread; LDS tree reduce.
// ---------------------------------------------------------------------------
__global__ __launch_bounds__(256) void k_molred(const float* __restrict__ a,
                                                const float* __restrict__ xyz,
                                                float* __restrict__ colsum,
                                                float* __restrict__ cg_xyz)
{
    __shared__ float red[256 * 32];               // 32 KB
    const int mol = blockIdx.x;
    const int t   = threadIdx.x;
    float p[32];
#pragma unroll
    for (int j = 0; j < 32; ++j) p[j] = 0.f;

    for (int i = t; i < N_PER; i += 256) {
        int atom = mol * N_PER + i;
        float x = xyz[atom * 3 + 0];
        float y = xyz[atom * 3 + 1];
        float z = xyz[atom * 3 + 2];
        const float* ar = a + (size_t)atom * N_CGS;
#pragma unroll
        for (int c = 0; c < N_CGS; ++c) {
            float av = ar[c];
            p[c]              += av;
            p[8 + c * 3 + 0]  += av * x;
            p[8 + c * 3 + 1]  += av * y;
            p[8 + c * 3 + 2]  += av * z;
        }
    }
#pragma unroll
    for (int j = 0; j < 32; ++j) red[t * 32 + j] = p[j];
    __syncthreads();
    for (int s = 128; s > 0; s >>= 1) {
        if (t < s)
#pragma unroll
            for (int j = 0; j < 32; ++j)
                red[t * 32 + j] += red[(t + s) * 32 + j];
        __syncthreads();
    }
    if (t == 0) {
#pragma unroll
        for (int c = 0; c < N_CGS; ++c) {
            float cs  = red[c];
            float inv = 1.f / cs;
            colsum[mol * N_CGS + c] = cs;
            cg_xyz[(mol * N_CGS + c) * 3 + 0] = red[8 + c * 3 + 0] * inv;
            cg_xyz[(mol * N_CGS + c) * 3 + 1] = red[8 + c * 3 + 1] * inv;
            cg_xyz[(mol * N_CGS + c) * 3 + 2] = red[8 + c * 3 + 2] * inv;
        }
    }
}

// ---------------------------------------------------------------------------
// Dense a_norm writer (single pass, zeros + the 8 live entries per row).
// thread = (atom, 4-col chunk of the 1024-wide row)
// ---------------------------------------------------------------------------
__global__ __launch_bounds__(256) void k_anorm(const float* __restrict__ a,
                                               const int* __restrict__ mol_id,
                                               const float* __restrict__ colsum,
                                               float* __restrict__ anorm)
{
    int tid  = blockIdx.x * 256 + threadIdx.x;
    int atom = tid >> 8;
    int cb   = (tid & 255) * 4;
    if (atom >= N_ATOMS) return;
    int base = mol_id[atom] * N_CGS;
    float4 v = make_float4(0.f, 0.f, 0.f, 0.f);
    float* vp = &v.x;
#pragma unroll
    for (int i = 0; i < 4; ++i) {
        int c = cb + i;
        if (c >= base && c < base + N_CGS)
            vp[i] = a[(size_t)atom * N_CGS + (c - base)] / colsum[c];
    }
    *(float4*)(anorm + (size_t)atom * BC + cb) = v;
}

// ---------------------------------------------------------------------------
// Zero-fill (float4 granular)
// ---------------------------------------------------------------------------
__global__ __launch_bounds__(256) void k_zero4(float* __restrict__ p, int n4)
{
    int tid = blockIdx.x * 256 + threadIdx.x;
    if (tid < n4) ((float4*)p)[tid] = make_float4(0.f, 0.f, 0.f, 0.f);
}

// ---------------------------------------------------------------------------
// cg_adj: per edge (p,q), 8x8 outer product of a[p],a[q] scattered into the
// (mol(p), mol(q)) 8x8 block with fp32 atomics. thread = (edge, ci).
// ---------------------------------------------------------------------------
__global__ __launch_bounds__(256) void k_adj(const int* __restrict__ edge,
                                             const float* __restrict__ a,
                                             const int* __restrict__ mol_id,
                                             float* __restrict__ adj)
{
    int tid = blockIdx.x * 256 + threadIdx.x;
    if (tid >= E_EDGE * N_CGS) return;
    int e  = tid >> 3;
    int ci = tid & 7;
    int p = edge[2 * e], q = edge[2 * e + 1];
    int mp = mol_id[p],  mq = mol_id[q];
    float ap = a[(size_t)p * N_CGS + ci];
    const float* aq = a + (size_t)q * N_CGS;
    float* row = adj + (size_t)(mp * N_CGS + ci) * BC + mq * N_CGS;
#pragma unroll
    for (int cj = 0; cj < N_CGS; ++cj)
        atomicAdd(row + cj, ap * aq[cj]);
}

// ---------------------------------------------------------------------------
// Launch
// ---------------------------------------------------------------------------
extern "C" void kernel_launch(void* const* d_in, const int* in_sizes, int n_in,
                              void* d_out, int out_size, void* d_ws, size_t ws_size,
                              hipStream_t stream)
{
    const int*   atoms = (const int*)  d_in[0];
    const float* xyz   = (const float*)d_in[1];
    const int*   bonds = (const int*)  d_in[2];
    const int*   edge  = (const int*)  d_in[3];
    const int*   mol   = (const int*)  d_in[4];
    const float* embed = (const float*)d_in[5];
    const float* Wu1   = (const float*)d_in[6];
    const float* bu1   = (const float*)d_in[7];
    const float* Wu2   = (const float*)d_in[8];
    const float* bu2   = (const float*)d_in[9];
    const float* Wc1   = (const float*)d_in[10];
    const float* bc1   = (const float*)d_in[11];
    const float* Wc2   = (const float*)d_in[12];
    const float* bc2   = (const float*)d_in[13];

    float* out    = (float*)d_out;
    float* hl     = out;                                    // 65536*8
    float* cg_xyz = out + (size_t)N_ATOMS * N_CGS;          // 1024*3
    float* anorm  = cg_xyz + (size_t)BC * 3;                // 65536*1024
    float* adj    = anorm + (size_t)N_ATOMS * BC;           // 1024*1024

    float* H  = (float*)d_ws;                               // N_ATOMS*128
    float* T  = H + (size_t)N_ATOMS * DIM;                  // N_ATOMS*128
    float* A  = T + (size_t)N_ATOMS * DIM;                  // N_ATOMS*8
    float* CS = A + (size_t)N_ATOMS * N_CGS;                // 1024

    // 1) h = embed[nodes] (row 0 zeroed)
    k_gather<<<(N_ATOMS * 32) / 256, 256, 0, stream>>>(atoms, embed, H);

    // 2) 3 conv layers, node-factored:  g = tanh(h@Wu1+b1)@Wu2+b2 ; h += Adj·g
    const int gemm_blocks = N_ATOMS / 128;                  // 8 waves * 16 rows
    for (int l = 0; l < N_CONV; ++l) {
        k_gemm128<<<gemm_blocks, 256, 0, stream>>>(
            H, Wu1 + (size_t)l * DIM * DIM, bu1 + (size_t)l * DIM, T, N_ATOMS, 1);
        k_gemm128<<<gemm_blocks, 256, 0, stream>>>(
            T, Wu2 + (size_t)l * DIM * DIM, bu2 + (size_t)l * DIM, T, N_ATOMS, 0); // in place
        k_scatter<<<(2 * E_BOND * 32) / 256, 256, 0, stream>>>(bonds, T, H);
    }

    // 3) classifier: T = tanh(H@Wc1+bc1); hl = T@Wc2+bc2; a = softmax(hl)
    k_gemm128<<<gemm_blocks, 256, 0, stream>>>(H, Wc1, bc1, T, N_ATOMS, 1);
    k_head<<<N_ATOMS / 256, 256, 0, stream>>>(T, Wc2, bc2, hl, A);

    // 4) per-mol colsum + cg_xyz (deterministic reduction)
    k_molred<<<N_MOLS, 256, 0, stream>>>(A, xyz, CS, cg_xyz);

    // 5) dense a_norm (zeros + scaled live entries, single pass)
    k_anorm<<<(N_ATOMS * 256) / 256, 256, 0, stream>>>(A, mol, CS, anorm);

    // 6) cg_adj = zero, then per-edge 8x8 outer-product atomics
    k_zero4<<<(BC * BC / 4) / 256, 256, 0, stream>>>(adj, BC * BC / 4);
    k_adj<<<(E_EDGE * N_CGS) / 256, 256, 0, stream>>>(edge, A, mol, adj);
}